// CostVolume2D_22531398435045
// MI455X (gfx1250) — compile-verified
//
#include <hip/hip_runtime.h>

// CostVolume2D for MI455X (gfx1250, wave32).
// B=8, H=256, W=512, C=32, n_disp=12 (fixed by the reference harness).
//
// Strategy: memory-minimal (each input byte read once), async global->LDS
// transposing copies (CDNA5 ASYNCcnt path), conflict-free LDS layout,
// VALU at the inherent 2-ops/element floor. This revision manually unrolls
// the async issue loops (the compiler declines to unroll loops containing
// the convergent async intrinsic), cutting per-column issue overhead.

#define BDIM   64      // 2 waves per block
#define TW     128     // output columns per block
#define ND     12      // disparities
#define CCH    32      // channels
#define WIDTH  512
#define HEIGHT 256
#define NB     8
#define LSTR   130     // sL stride in dwords: even, ==2 mod 4 -> conflict-free
#define RSTR   142     // sR stride in dwords: even, ==2 mod 4 -> conflict-free
#define RCOLS  140     // r columns loaded: 12 halo + 128

#define AS1 __attribute__((address_space(1)))
#define AS3 __attribute__((address_space(3)))

// ---- CDNA5 async global->LDS copy (one dword per lane, DMA-side LDS write).
// One wave-instruction moves one tensor column: 32 channels, 128 contiguous
// bytes from global, scatter-transposed into channel-major LDS (each lane
// carries its own LDS destination address -> the transpose is free).
__device__ __forceinline__ void async_copy_b32(const float* g, float* l) {
#if __has_builtin(__builtin_amdgcn_global_load_async_to_lds_b32)
  __builtin_amdgcn_global_load_async_to_lds_b32(
      (AS1 int*)(unsigned long long)(size_t)g,
      (AS3 int*)(unsigned)(size_t)l,
      /*imm offset*/ 0, /*cpol*/ 0);
#else
  unsigned loff = (unsigned)(size_t)l;
  asm volatile("global_load_async_to_lds_b32 %0, %1, off"
               :: "v"(loff), "v"(g) : "memory");
#endif
}

__device__ __forceinline__ void wait_async0() {
#if __has_builtin(__builtin_amdgcn_s_wait_asynccnt)
  __builtin_amdgcn_s_wait_asynccnt(0);
#else
  asm volatile("s_wait_asynccnt 0" ::: "memory");
#endif
}

__global__ __launch_bounds__(BDIM)
void cost_volume_kernel(const float* __restrict__ feat_l,
                        const float* __restrict__ feat_r,
                        float* __restrict__ out) {
  // Channel-major LDS tiles sX[c][j]. Strides == 2 (mod 4), even:
  //  * async scatter stores (lane==c) hit 32 distinct banks,
  //  * compute-phase float2 reads are 8B-aligned and cover 64 consecutive
  //    dwords per wave -> every bank exactly once.
  __shared__ float sL[CCH * LSTR];   // 16,640 B
  __shared__ float sR[CCH * RSTR];   // 18,176 B

  const int gid   = blockIdx.x;
  const int tile  = gid & 3;         // 4 tiles of 128 columns per row
  const int bh    = gid >> 2;        // 0 .. B*H-1
  const int wbase = tile * TW;       // SGPR-uniform

  const int t    = threadIdx.x;      // 0..63
  const int lane = t & 31;           // channel index during load phase
  const int wv   = t >> 5;           // wave id 0..1

  const float* gL = feat_l + (size_t)bh * WIDTH * CCH + lane;
  const float* gR = feat_r + (size_t)bh * WIDTH * CCH + lane;
  float* sRrow = &sR[lane * RSTR];
  float* sLrow = &sL[lane * LSTR];

  // ---------------- load phase (uniform trips, manual unroll) -------------
  // Each wave covers every other column starting at its wave id.
  if (wbase == 0) {                        // scalar-uniform branch
    // reference zero-pads R on the left: halo columns 0..11 are zeros
#pragma unroll
    for (int k = 0; k < 6; ++k)
      sRrow[wv + 2 * k] = 0.0f;
    // r columns 12..139  ->  gj = col-12 in 0..127  (64 cols/wave, 4-wide)
    const float* g = gR + (size_t)wv * CCH;
    float*       s = sRrow + 12 + wv;
    for (int k = 0; k < 16; ++k) {
      async_copy_b32(g,           s);
      async_copy_b32(g + 2 * CCH, s + 2);
      async_copy_b32(g + 4 * CCH, s + 4);
      async_copy_b32(g + 6 * CCH, s + 6);
      g += 8 * CCH; s += 8;
    }
  } else {
    // r columns 0..139, all in-bounds (70 cols/wave, 5-wide)
    const float* g = gR + (size_t)(wbase - 12 + wv) * CCH;
    float*       s = sRrow + wv;
    for (int k = 0; k < 14; ++k) {
      async_copy_b32(g,           s);
      async_copy_b32(g + 2 * CCH, s + 2);
      async_copy_b32(g + 4 * CCH, s + 4);
      async_copy_b32(g + 6 * CCH, s + 6);
      async_copy_b32(g + 8 * CCH, s + 8);
      g += 10 * CCH; s += 10;
    }
  }
  {
    // l columns 0..127 (64 cols/wave, 4-wide)
    const float* g = gL + (size_t)(wbase + wv) * CCH;
    float*       s = sLrow + wv;
    for (int k = 0; k < 16; ++k) {
      async_copy_b32(g,           s);
      async_copy_b32(g + 2 * CCH, s + 2);
      async_copy_b32(g + 4 * CCH, s + 4);
      async_copy_b32(g + 6 * CCH, s + 6);
      g += 8 * CCH; s += 8;
    }
  }

  wait_async0();
  __syncthreads();

  // ---------------- compute phase ----------------
  // Thread t owns local output columns wl0 = 2t and wl0+1.
  // Disparity d needs r column j = wl + 12 - d  ->  j in [wl0, wl0+13].
  const int wl0 = t << 1;

  float acc0[ND], acc1[ND];
#pragma unroll
  for (int d = 0; d < ND; ++d) { acc0[d] = 0.0f; acc1[d] = 0.0f; }

#pragma unroll 4
  for (int c = 0; c < CCH; ++c) {
    const float2* rp = (const float2*)&sR[c * RSTR + wl0];  // 8B aligned
    float rw[14];
#pragma unroll
    for (int m = 0; m < 7; ++m) {          // 7x ds_load_b64, conflict-free
      float2 v = rp[m];
      rw[2 * m]     = v.x;
      rw[2 * m + 1] = v.y;
    }
    const float2 lv = *(const float2*)&sL[c * LSTR + wl0];
#pragma unroll
    for (int d = 0; d < ND; ++d) {
      // v_sub_f32 + v_add_f32 with |src| modifier: 2 VALU per element.
      acc0[d] += __builtin_fabsf(lv.x - rw[12 - d]);
      acc1[d] += __builtin_fabsf(lv.y - rw[13 - d]);
    }
  }

  // ---------------- store phase ----------------
  // out[b,h,w,d]: 12 floats per column, 48B column stride -> 16B aligned.
  float* op = out + ((size_t)bh * WIDTH + wbase + wl0) * ND;
  float4* o4 = (float4*)op;
  o4[0] = make_float4(acc0[0], acc0[1], acc0[2],  acc0[3]);
  o4[1] = make_float4(acc0[4], acc0[5], acc0[6],  acc0[7]);
  o4[2] = make_float4(acc0[8], acc0[9], acc0[10], acc0[11]);
  o4[3] = make_float4(acc1[0], acc1[1], acc1[2],  acc1[3]);
  o4[4] = make_float4(acc1[4], acc1[5], acc1[6],  acc1[7]);
  o4[5] = make_float4(acc1[8], acc1[9], acc1[10], acc1[11]);
}

extern "C" void kernel_launch(void* const* d_in, const int* in_sizes, int n_in,
                              void* d_out, int out_size, void* d_ws, size_t ws_size,
                              hipStream_t stream) {
  (void)in_sizes; (void)n_in; (void)d_ws; (void)ws_size; (void)out_size;
  const float* feat_l = (const float*)d_in[0];
  const float* feat_r = (const float*)d_in[1];
  float* out = (float*)d_out;

  const int blocks = NB * HEIGHT * (WIDTH / TW);   // 8 * 256 * 4 = 8192
  cost_volume_kernel<<<dim3(blocks), dim3(BDIM), 0, stream>>>(feat_l, feat_r, out);
}